// KLayerSAGE_30133490549162
// MI455X (gfx1250) — compile-verified
//
#include <hip/hip_runtime.h>

typedef __attribute__((ext_vector_type(2))) float v2f;
typedef __attribute__((ext_vector_type(8))) float v8f;

#define HD 128
#define GG 128
#define CC 10

__device__ __forceinline__ float atomAddF(float* p, float v) {
  return __hip_atomic_fetch_add(p, v, __ATOMIC_RELAXED, __HIP_MEMORY_SCOPE_AGENT);
}

__device__ __forceinline__ v8f wmma_f32(v2f a, v2f b, v8f c) {
  // V_WMMA_F32_16X16X4_F32: D = A(16x4 f32) x B(4x16 f32) + C(16x16 f32)
  return __builtin_amdgcn_wmma_f32_16x16x4_f32(false, a, false, b, (short)0, c,
                                               false, false);
}

__global__ void zero_kernel(float* __restrict__ p, long n) {
  long i = (long)blockIdx.x * blockDim.x + threadIdx.x;
  const long stride = (long)gridDim.x * blockDim.x;
  for (; i < n; i += stride) p[i] = 0.f;
}

__global__ void deg_kernel(const int* __restrict__ dst, float* __restrict__ deg, int E) {
  int i = blockIdx.x * blockDim.x + threadIdx.x;
  const int stride = gridDim.x * blockDim.x;
  for (; i < E; i += stride) atomAddF(&deg[dst[i]], 1.f);
}

__global__ void invdeg_kernel(float* __restrict__ deg, int N) {
  int i = blockIdx.x * blockDim.x + threadIdx.x;
  const int stride = gridDim.x * blockDim.x;
  for (; i < N; i += stride) deg[i] = 1.f / fmaxf(deg[i], 1.f);
}

__global__ void cnt_kernel(const int* __restrict__ ng, float* __restrict__ cnt, int N) {
  int i = blockIdx.x * blockDim.x + threadIdx.x;
  const int stride = gridDim.x * blockDim.x;
  for (; i < N; i += stride) atomAddF(&cnt[ng[i]], 1.f);
}

// one wave per edge; lane q handles features [4q, 4q+4)
__global__ void scatter_kernel(const float4* __restrict__ feat,
                               const int* __restrict__ src,
                               const int* __restrict__ dst,
                               float* __restrict__ acc, long total) {
  long i = (long)blockIdx.x * blockDim.x + threadIdx.x;
  const long stride = (long)gridDim.x * blockDim.x;
  for (; i < total; i += stride) {
    const int e = (int)(i >> 5);        // wave-uniform
    const int q = (int)(i & 31);
    const float4 v = feat[(long)src[e] * (HD / 4) + q];
    float* o = acc + (long)dst[e] * HD + q * 4;
    atomAddF(o + 0, v.x);
    atomAddF(o + 1, v.y);
    atomAddF(o + 2, v.z);
    atomAddF(o + 3, v.w);
  }
}

// Y = leaky_relu(elu(X @ Ws^T + (NB * inv_deg) @ Wn^T + b))
// FUSE_POOL: atomically accumulate rows into pooled[node_graph[row]] instead of storing Y.
template <bool FUSE_POOL>
__launch_bounds__(256, 2)
__global__ void sage_layer_kernel(const float* __restrict__ X,
                                  const float* __restrict__ NB,
                                  const float* __restrict__ invdeg,
                                  const float* __restrict__ Ws,
                                  const float* __restrict__ Wn,
                                  const float* __restrict__ bias,
                                  float* __restrict__ Y,
                                  const int* __restrict__ node_graph,
                                  float* __restrict__ pooled,
                                  int N, int nRowTiles) {
  __shared__ float sWs[HD * HD];  // 64 KB
  __shared__ float sWn[HD * HD];  // 64 KB
  {
    const float4* a = (const float4*)Ws;
    const float4* b = (const float4*)Wn;
    float4* sa = (float4*)sWs;
    float4* sb = (float4*)sWn;
    for (int i = threadIdx.x; i < HD * HD / 4; i += blockDim.x) {
      sa[i] = a[i];
      sb[i] = b[i];
    }
  }
  __syncthreads();

  const int lane = threadIdx.x & 31;
  const int wave = threadIdx.x >> 5;  // column tile 0..7
  const int m = lane & 15;
  const int kk = lane >> 4;           // A frag: lanes 0-15 hold K{0,1}, 16-31 hold K{2,3}
  const int col = wave * 16 + m;      // B/C column owned by this lane
  const float* wsrow = sWs + col * HD;
  const float* wnrow = sWn + col * HD;
  const float bcol = bias[col];

  for (int rt = blockIdx.x; rt < nRowTiles; rt += gridDim.x) {
    const int row0 = rt * 16;
    int arow = row0 + m;
    if (arow >= N) arow = N - 1;      // clamp (N % 16 == 0 here, guard is free)
    const float* xrow = X + (long)arow * HD;
    const float* nrow = NB + (long)arow * HD;
    const float idg = invdeg[arow];

    v8f acc = {0.f, 0.f, 0.f, 0.f, 0.f, 0.f, 0.f, 0.f};
#pragma unroll 8
    for (int k = 0; k < HD; k += 4) {
      const int ka = k + 2 * kk;
      v2f a0, b0, a1, b1;
      a0[0] = xrow[ka];
      a0[1] = xrow[ka + 1];
      b0[0] = wsrow[ka];
      b0[1] = wsrow[ka + 1];
      acc = wmma_f32(a0, b0, acc);
      a1[0] = nrow[ka] * idg;
      a1[1] = nrow[ka + 1] * idg;
      b1[0] = wnrow[ka];
      b1[1] = wnrow[ka + 1];
      acc = wmma_f32(a1, b1, acc);
    }

    // C/D layout: lane holds col = lane%16, rows v + 8*(lane/16)
    const int rbase = row0 + 8 * kk;
#pragma unroll
    for (int v = 0; v < 8; ++v) {
      const int r = rbase + v;
      if (r < N) {
        float y = acc[v] + bcol;
        y = (y > 0.f) ? y : (__expf(y) - 1.f);  // elu (alpha=1)
        y = (y >= 0.f) ? y : 0.01f * y;         // leaky_relu
        if (FUSE_POOL) {
          atomAddF(&pooled[(long)node_graph[r] * HD + col], y);
        } else {
          Y[(long)r * HD + col] = y;
        }
      }
    }
  }
}

// per-graph: hg = pooled/cnt ; hl = leaky(hg@Wl^T + bl) ; out = hl@Wo^T + bo
__global__ void readout_kernel(const float* __restrict__ pooled,
                               const float* __restrict__ cnt,
                               const float* __restrict__ Wl,
                               const float* __restrict__ bl,
                               const float* __restrict__ Wo,
                               const float* __restrict__ bo,
                               float* __restrict__ out) {
  const int g = blockIdx.x;
  const int t = threadIdx.x;
  __shared__ float hg[HD];
  __shared__ float hl[HD];
  const float c = fmaxf(cnt[g], 1.f);
  hg[t] = pooled[(long)g * HD + t] / c;
  __syncthreads();
  float s = bl[t];
  const float* wrow = Wl + (long)t * HD;
  for (int k = 0; k < HD; ++k) s += hg[k] * wrow[k];
  hl[t] = (s >= 0.f) ? s : 0.01f * s;
  __syncthreads();
  if (t < CC) {
    float o = bo[t];
    const float* worow = Wo + (long)t * HD;
    for (int k = 0; k < HD; ++k) o += hl[k] * worow[k];
    out[(long)g * CC + t] = o;
  }
}

extern "C" void kernel_launch(void* const* d_in, const int* in_sizes, int n_in,
                              void* d_out, int out_size, void* d_ws, size_t ws_size,
                              hipStream_t stream) {
  const float* x = (const float*)d_in[0];
  const int* src = (const int*)d_in[1];
  const int* dst = (const int*)d_in[2];
  const int* ng = (const int*)d_in[3];
  const float* Ws0 = (const float*)d_in[4];
  const float* Wn0 = (const float*)d_in[5];
  const float* b0 = (const float*)d_in[6];
  const float* Ws1 = (const float*)d_in[7];
  const float* Wn1 = (const float*)d_in[8];
  const float* b1 = (const float*)d_in[9];
  const float* Wl = (const float*)d_in[10];
  const float* bl = (const float*)d_in[11];
  const float* Wo = (const float*)d_in[12];
  const float* bo = (const float*)d_in[13];

  const int N = in_sizes[0] / HD;
  const int E = in_sizes[1];
  if (N <= 0) return;

  float* ws = (float*)d_ws;
  float* deg = ws;                      // N (becomes inv_deg in place)
  float* pooled = deg + N;              // G*H
  float* cnt = pooled + (long)GG * HD;  // G
  float* neigh = cnt + GG;              // N*H
  float* hbuf = neigh + (long)N * HD;   // N*H

  const long zero_n = (long)N + (long)GG * HD + GG + (long)N * HD;
  const int nRowTiles = (N + 15) / 16;

  zero_kernel<<<2048, 256, 0, stream>>>(ws, zero_n);
  deg_kernel<<<1024, 256, 0, stream>>>(dst, deg, E);
  invdeg_kernel<<<512, 256, 0, stream>>>(deg, N);
  cnt_kernel<<<512, 256, 0, stream>>>(ng, cnt, N);

  // layer 0
  scatter_kernel<<<4096, 256, 0, stream>>>((const float4*)x, src, dst, neigh,
                                           (long)E * 32);
  sage_layer_kernel<false><<<1024, 256, 0, stream>>>(
      x, neigh, deg, Ws0, Wn0, b0, hbuf, nullptr, nullptr, N, nRowTiles);

  // layer 1 (fused mean-pool numerator)
  zero_kernel<<<2048, 256, 0, stream>>>(neigh, (long)N * HD);
  scatter_kernel<<<4096, 256, 0, stream>>>((const float4*)hbuf, src, dst, neigh,
                                           (long)E * 32);
  sage_layer_kernel<true><<<1024, 256, 0, stream>>>(
      hbuf, neigh, deg, Ws1, Wn1, b1, nullptr, ng, pooled, N, nRowTiles);

  // readout MLP
  readout_kernel<<<GG, HD, 0, stream>>>(pooled, cnt, Wl, bl, Wo, bo,
                                        (float*)d_out);
}